// GraphConvolution_21869973471659
// MI455X (gfx1250) — compile-verified
//
#include <hip/hip_runtime.h>

typedef float v2f __attribute__((ext_vector_type(2)));
typedef float v8f __attribute__((ext_vector_type(8)));

#define D_FEAT 128
#define N_TILES 8   // 128 / 16 column tiles

// -------------------------------------------------------------------------
// Kernel 1: support = X @ W using V_WMMA_F32_16X16X4_F32 (full f32 precision).
// One wave computes a 16(M) x 128(N) strip: 8 accumulators, K-loop in steps
// of 4. A-fragment layout (ISA 32-bit A 16x4): lanes 0-15 hold M=0..15 with
// v0=K0,v1=K1; lanes 16-31 hold v0=K2,v1=K3. B (4x16) mirrors this with
// N striped across lanes. C/D: VGPR r -> row r (lanes 0-15) / row r+8
// (lanes 16-31), N = lane&15.
// -------------------------------------------------------------------------
__global__ __launch_bounds__(128) void gcn_gemm_wmma(
    const float* __restrict__ X, const float* __restrict__ W,
    float* __restrict__ S, int nRows)
{
  const int lane  = threadIdx.x & 31;
  const int wave  = threadIdx.x >> 5;
  const int tile  = blockIdx.x * 4 + wave;   // wave-uniform
  const int row0  = tile * 16;
  if (row0 >= nRows) return;                 // uniform exit: EXEC stays all-1 for WMMA

  const int mrow  = lane & 15;
  const int khalf = lane >> 4;               // 0 or 1

  v8f acc[N_TILES];
  #pragma unroll
  for (int nt = 0; nt < N_TILES; ++nt) acc[nt] = (v8f)(0.0f);

  // clamp A row reads for a ragged tail (unused here: 100000 % 16 == 0)
  int arowi = row0 + mrow;
  if (arowi >= nRows) arowi = nRows - 1;
  const float* arow = X + (size_t)arowi * D_FEAT;

  for (int k0 = 0; k0 < D_FEAT; k0 += 4) {
    const int kk = k0 + 2 * khalf;                       // even -> 8B aligned
    const v2f a = *reinterpret_cast<const v2f*>(arow + kk);
    const float* Wk = W + (size_t)kk * D_FEAT;
    #pragma unroll
    for (int nt = 0; nt < N_TILES; ++nt) {
      const int col = nt * 16 + mrow;
      v2f b;
      b.x = Wk[col];            // K = kk
      b.y = Wk[col + D_FEAT];   // K = kk + 1
      acc[nt] = __builtin_amdgcn_wmma_f32_16x16x4_f32(
          false, a, false, b, (short)0, acc[nt], false, false);
    }
  }

  // Store: single wave-uniform full-tile test instead of per-element guards
  // (removes 64x v_cmp + s_and_saveexec in the tail; full path is the only
  // one taken for N_NODES = 100000).
  float* Sbase = S + (size_t)row0 * D_FEAT + (size_t)(8 * khalf) * D_FEAT + mrow;
  if (row0 + 16 <= nRows) {
    #pragma unroll
    for (int nt = 0; nt < N_TILES; ++nt) {
      #pragma unroll
      for (int r = 0; r < 8; ++r) {
        Sbase[(size_t)r * D_FEAT + nt * 16] = acc[nt][r];
      }
    }
  } else {
    #pragma unroll
    for (int nt = 0; nt < N_TILES; ++nt) {
      #pragma unroll
      for (int r = 0; r < 8; ++r) {
        const int row = row0 + r + 8 * khalf;
        if (row < nRows) S[(size_t)row * D_FEAT + nt * 16 + mrow] = acc[nt][r];
      }
    }
  }
}

// -------------------------------------------------------------------------
// Kernel 2: out[i] = bias[i % 128]  (d_out is poisoned by the harness)
// -------------------------------------------------------------------------
__global__ __launch_bounds__(256) void gcn_init_bias(
    const float* __restrict__ bias, float* __restrict__ out, int total)
{
  const int i = blockIdx.x * blockDim.x + threadIdx.x;
  if (i < total) out[i] = bias[i & (D_FEAT - 1)];
}

// -------------------------------------------------------------------------
// Kernel 3: COO SpMM scatter. One wave per edge: float4 (B128) gather of the
// support row (L2-resident: 51.2 MB < 192 MB L2), scale by edge weight,
// 4x global_atomic_add_f32 per lane into out[dst] (also L2-resident).
// -------------------------------------------------------------------------
__global__ __launch_bounds__(256) void gcn_scatter(
    const float* __restrict__ S, const int* __restrict__ ei,
    const float* __restrict__ ew, float* __restrict__ out, int nEdges)
{
  const long long gid  = (long long)blockIdx.x * blockDim.x + threadIdx.x;
  const long long edge = gid >> 5;
  const int lane       = (int)(gid & 31);
  if (edge >= nEdges) return;

  const int dst = ei[edge];                       // edge_index[0]
  const int src = ei[(long long)nEdges + edge];   // edge_index[1]
  const float w = ew[edge];

  const float4* srow = reinterpret_cast<const float4*>(S + (size_t)src * D_FEAT);
  float*        drow = out + (size_t)dst * D_FEAT;

  const float4 v = srow[lane];         // 128 floats = 32 lanes x float4
  atomicAdd(drow + lane * 4 + 0, v.x * w);
  atomicAdd(drow + lane * 4 + 1, v.y * w);
  atomicAdd(drow + lane * 4 + 2, v.z * w);
  atomicAdd(drow + lane * 4 + 3, v.w * w);
}

// -------------------------------------------------------------------------
// Inputs (setup_inputs order):
//   d_in[0] x           float32 [100000*128]
//   d_in[1] weight      float32 [128*128]
//   d_in[2] bias        float32 [128]
//   d_in[3] edge_index  int32   [2*1600000]  (row0 = dst, row1 = src)
//   d_in[4] edge_weight float32 [1600000]
// d_ws: support buffer, needs N_NODES*128*4 = 51.2 MB
// -------------------------------------------------------------------------
extern "C" void kernel_launch(void* const* d_in, const int* in_sizes, int n_in,
                              void* d_out, int out_size, void* d_ws, size_t ws_size,
                              hipStream_t stream) {
  const float* x  = (const float*)d_in[0];
  const float* w  = (const float*)d_in[1];
  const float* b  = (const float*)d_in[2];
  const int*   ei = (const int*)d_in[3];
  const float* ew = (const float*)d_in[4];
  float* out = (float*)d_out;
  float* S   = (float*)d_ws;

  const int nNodes = in_sizes[0] / D_FEAT;   // 100000
  const int nEdges = in_sizes[4];            // 1600000

  // 1) support = X @ W  (WMMA f32)
  const int tiles = (nNodes + 15) / 16;      // 6250
  gcn_gemm_wmma<<<dim3((tiles + 3) / 4), dim3(128), 0, stream>>>(x, w, S, nNodes);

  // 2) out = bias (broadcast)
  const int total = nNodes * D_FEAT;
  gcn_init_bias<<<dim3((total + 255) / 256), dim3(256), 0, stream>>>(b, out, total);

  // 3) out += A @ support  (edge-parallel scatter, f32 atomics in L2)
  const long long threads = (long long)nEdges * 32;
  gcn_scatter<<<dim3((unsigned)((threads + 255) / 256)), dim3(256), 0, stream>>>(
      S, ei, ew, out, nEdges);
}